// RWKV_Tmix_x060_37864431682265
// MI455X (gfx1250) — compile-verified
//
#include <hip/hip_runtime.h>

typedef float v2f __attribute__((ext_vector_type(2)));
typedef float v8f __attribute__((ext_vector_type(8)));

#define Bc 32
#define Ec 2048
#define ROWSc 1584      // (2+S)*NL
#define I1ROW 793       // (2+S)*LID + 1
#define SROW 794        // start of s slice
#define Hc 32
#define Sc 64
#define KC 4            // split-K factor
#define KCHUNK 512      // 2048 / KC
#define EPSC 6.4e-4f    // 1e-5 * 8^2

// ---------------- bulk state passthrough (skips rows 793..857 per batch) ------
// rows 793 (token shift) and 794..857 (s_new) are fully rewritten by prep/att_gn,
// so skipping them here saves ~17MB of doubly-written HBM traffic.
__global__ void copy_state(const float4* __restrict__ src, float4* __restrict__ dst, long n4) {
    long i = (long)blockIdx.x * blockDim.x + threadIdx.x;
    long stride = (long)gridDim.x * blockDim.x;
    for (; i < n4; i += stride) {
        int rowg = (int)(i >> 9);            // 512 float4 per E-row
        int r = rowg - (rowg / ROWSc) * ROWSc;
        if (r >= I1ROW && r < SROW + Sc) continue;
        dst[i] = src[i];
    }
}

// ---------------- sx, xxx, and new_state[:,793,:] = x ----------------
__global__ void prep(const float* __restrict__ x, const float* __restrict__ state,
                     const float* __restrict__ maa_x,
                     float* __restrict__ sx, float* __restrict__ xxx,
                     float* __restrict__ new_state) {
    int idx = blockIdx.x * blockDim.x + threadIdx.x;   // 0..65535
    int b = idx >> 11, e = idx & 2047;
    size_t soff = (size_t)b * ROWSc * Ec + (size_t)I1ROW * Ec + e;
    float xv = x[idx];
    float s  = state[soff] - xv;
    sx[idx]  = s;
    xxx[idx] = xv + s * maa_x[e];
    new_state[soff] = xv;
}

// ---------------- T = tanh(xxx @ maa_w1)  [32 x 160] ----------------
__global__ void mix1(const float* __restrict__ xxx, const float* __restrict__ w1,
                     float* __restrict__ T) {
    int b = blockIdx.x, j = threadIdx.x;               // j < 160
    const float* xr = xxx + b * Ec;
    float acc = 0.f;
    for (int e = 0; e < Ec; ++e) acc += xr[e] * w1[e * 160 + j];
    T[b * 160 + j] = tanhf(acc);
}

// ---------------- m_i and xw/xr/xk/xv/xg ----------------
__global__ void mix2(const float* __restrict__ x, const float* __restrict__ sx,
                     const float* __restrict__ T, const float* __restrict__ w2,
                     const float* __restrict__ maa_w, const float* __restrict__ maa_k,
                     const float* __restrict__ maa_v, const float* __restrict__ maa_r,
                     const float* __restrict__ maa_g,
                     float* __restrict__ xw, float* __restrict__ xbuf) {
    int b = blockIdx.x;
    int e = blockIdx.y * blockDim.x + threadIdx.x;
    int idx = b * Ec + e;
    float m[5];
#pragma unroll
    for (int i = 0; i < 5; ++i) {
        // t.reshape(5,B,32) of the flat [B,160] matrix
        const float* tp = T + i * 1024 + b * 32;
        const float* wp = w2 + (size_t)(i * 32) * Ec + e;
        float acc = 0.f;
#pragma unroll
        for (int d = 0; d < 32; ++d) acc += tp[d] * wp[(size_t)d * Ec];
        m[i] = acc;
    }
    float xv = x[idx], s = sx[idx];
    xw[idx] = xv + s * (maa_w[e] + m[0]);
    xbuf[0 * Bc * Ec + idx] = xv + s * (maa_r[e] + m[3]);   // xr -> Wr
    xbuf[1 * Bc * Ec + idx] = xv + s * (maa_k[e] + m[1]);   // xk -> Wk
    xbuf[2 * Bc * Ec + idx] = xv + s * (maa_v[e] + m[2]);   // xv -> Wv
    xbuf[3 * Bc * Ec + idx] = xv + s * (maa_g[e] + m[4]);   // xg -> Wg
}

// ---------------- T2 = tanh(xw @ decay_w1)  [32 x 64] ----------------
__global__ void decay1(const float* __restrict__ xw, const float* __restrict__ dw1,
                       float* __restrict__ T2) {
    int b = blockIdx.x, d = threadIdx.x;               // d < 64
    const float* xr = xw + b * Ec;
    float acc = 0.f;
    for (int e = 0; e < Ec; ++e) acc += xr[e] * dw1[e * 64 + d];
    T2[b * 64 + d] = tanhf(acc);
}

// ---------------- w = exp(-exp(decay + T2 @ decay_w2)) ----------------
__global__ void decay2(const float* __restrict__ T2, const float* __restrict__ dw2,
                       const float* __restrict__ decay, float* __restrict__ Wd) {
    int b = blockIdx.x;
    int j = blockIdx.y * blockDim.x + threadIdx.x;
    const float* tp = T2 + b * 64;
    float acc = decay[j];
#pragma unroll 8
    for (int d = 0; d < 64; ++d) acc += tp[d] * dw2[(size_t)d * Ec + j];
    Wd[b * Ec + j] = expf(-expf(acc));
}

// ---------------- one-wave 32x16 output tile over a K chunk ------------------
// Y[b][n] = sum_e X[b][e] * W[n][e] for K in [kbeg, kend).
// fp32 WMMA lane map: A 16x4: (M=lane&15, Kpair=(lane>>4)*2); B 4x16 symmetric;
// D v8f: M = vgpr + (lane>>4)*8, N = lane&15.
__device__ __forceinline__ void gemm_tile(const float* __restrict__ X,
                                          const float* __restrict__ W,
                                          float* __restrict__ Y,
                                          int nt, int lane, int kbeg, int kend) {
    int row = lane & 15;
    int ks  = (lane >> 4) << 1;
    const float* xa = X + (size_t)row * Ec;
    const float* xb = X + (size_t)(row + 16) * Ec;
    const float* wp = W + (size_t)(nt * 16 + row) * Ec;
    v8f acc0, acc1;
#pragma unroll
    for (int j = 0; j < 8; ++j) { acc0[j] = 0.f; acc1[j] = 0.f; }
    for (int k0 = kbeg; k0 < kend; k0 += 64) {
        __builtin_prefetch(wp + k0 + 256, 0, 1);   // ~1KB ahead in streamed row
#pragma unroll
        for (int kk = 0; kk < 64; kk += 4) {
            int k = k0 + kk;
            v2f a0 = *(const v2f*)(xa + k + ks);
            v2f a1 = *(const v2f*)(xb + k + ks);
            v2f bb = *(const v2f*)(wp + k + ks);
            acc0 = __builtin_amdgcn_wmma_f32_16x16x4_f32(false, a0, false, bb,
                                                         (short)0, acc0, false, false);
            acc1 = __builtin_amdgcn_wmma_f32_16x16x4_f32(false, a1, false, bb,
                                                         (short)0, acc1, false, false);
        }
    }
    int col = nt * 16 + row;
    int mb  = (lane >> 4) * 8;
#pragma unroll
    for (int j = 0; j < 8; ++j) {
        Y[(size_t)(mb + j) * Ec + col]      = acc0[j];
        Y[(size_t)(16 + mb + j) * Ec + col] = acc1[j];
    }
}

// ---- fused r/k/v/g projections, split-K x4: 2048 waves in one launch --------
__global__ void gemm_rkvg_splitk(const float* __restrict__ xbuf,
                                 const float* __restrict__ Wr, const float* __restrict__ Wk,
                                 const float* __restrict__ Wv, const float* __restrict__ Wg,
                                 float* __restrict__ part) {
    int wid = blockIdx.x * (blockDim.x >> 5) + (threadIdx.x >> 5);  // 0..2047
    int lane = threadIdx.x & 31;
    int g  = wid >> 9;                 // gemm id 0..3 (wave-uniform)
    int rem = wid & 511;
    int nt = rem >> 2;                 // N tile 0..127
    int kc = rem & 3;                  // K chunk 0..3
    const float* X = xbuf + (size_t)g * Bc * Ec;
    const float* W = (g == 0) ? Wr : (g == 1) ? Wk : (g == 2) ? Wv : Wg;
    float* P = part + (size_t)(g * KC + kc) * Bc * Ec;
    gemm_tile(X, W, P, nt, lane, kc * KCHUNK, kc * KCHUNK + KCHUNK);
}

// ---- single GEMM, split-K x4: 512 waves ------------------------------------
__global__ void gemm_one_splitk(const float* __restrict__ X, const float* __restrict__ W,
                                float* __restrict__ part) {
    int wid = blockIdx.x * (blockDim.x >> 5) + (threadIdx.x >> 5);  // 0..511
    int lane = threadIdx.x & 31;
    int nt = wid >> 2, kc = wid & 3;
    gemm_tile(X, W, part + (size_t)kc * Bc * Ec, nt, lane, kc * KCHUNK, kc * KCHUNK + KCHUNK);
}

// ---- deterministic split-K reduction: out[idx] = sum of 4 partials ----------
__global__ void reduce_k4(const float* __restrict__ part, float* __restrict__ out) {
    int idx = blockIdx.x * blockDim.x + threadIdx.x;
    int g = idx >> 16, r = idx & 65535;
    const float* p = part + (size_t)(g * KC) * Bc * Ec + r;
    out[idx] = ((p[0] + p[1 * Bc * Ec]) + (p[2 * Bc * Ec] + p[3 * Bc * Ec]));
}

// ---------------- per-(b,h) attention + s_new + GroupNorm + SiLU gate --------
__global__ void att_gn(const float* __restrict__ RKVG, const float* __restrict__ Wd,
                       const float* __restrict__ faaaa, const float* __restrict__ state,
                       const float* __restrict__ gnw, const float* __restrict__ gnb,
                       float* __restrict__ new_state, float* __restrict__ G) {
    __shared__ float sh_r[64], sh_k[64], sh_w[64], sh_rfk[64], sh_o[64];
    int bh = blockIdx.x, b = bh >> 5, h = bh & 31;
    int j  = threadIdx.x;                   // 0..63 (output column)
    int ch = h * 64 + j;
    const float* R  = RKVG + 0 * Bc * Ec;
    const float* K  = RKVG + 1 * Bc * Ec;
    const float* V  = RKVG + 2 * Bc * Ec;
    const float* Gp = RKVG + 3 * Bc * Ec;
    float rj = R[b * Ec + ch], kj = K[b * Ec + ch];
    float vj = V[b * Ec + ch], wj = Wd[b * Ec + ch];
    sh_r[j] = rj; sh_k[j] = kj; sh_w[j] = wj;
    sh_rfk[j] = rj * faaaa[ch] * kj;
    __syncthreads();
    float c = 0.f;
#pragma unroll 8
    for (int i = 0; i < 64; ++i) c += sh_rfk[i];
    size_t soff = (size_t)b * ROWSc * Ec + (size_t)SROW * Ec + (size_t)h * 4096;
    const float* sp = state + soff;
    float* np = new_state + soff;
    float o = c * vj;                        // o_j = (Σ r_i f_i k_i) v_j + Σ r_i s_ij
#pragma unroll 4
    for (int i = 0; i < 64; ++i) {
        float sv = sp[i * 64 + j];
        o += sh_r[i] * sv;
        np[i * 64 + j] = sh_k[i] * vj + sh_w[i] * sv;   // s_new = k⊗v + w*s
    }
    sh_o[j] = o;
    __syncthreads();
    float mu = 0.f, m2 = 0.f;
    for (int i = 0; i < 64; ++i) mu += sh_o[i];
    mu *= (1.f / 64.f);
    for (int i = 0; i < 64; ++i) { float d = sh_o[i] - mu; m2 += d * d; }
    float var = m2 * (1.f / 64.f);
    float normed = (o - mu) * rsqrtf(var + EPSC) * gnw[ch] + gnb[ch];
    float gp = Gp[b * Ec + ch];
    G[b * Ec + ch] = normed * (gp / (1.f + expf(-gp)));   // SiLU gate
}

extern "C" void kernel_launch(void* const* d_in, const int* in_sizes, int n_in,
                              void* d_out, int out_size, void* d_ws, size_t ws_size,
                              hipStream_t stream) {
    const float* x     = (const float*)d_in[0];
    const float* state = (const float*)d_in[1];
    const float* maa_x = (const float*)d_in[2];
    const float* maa_w = (const float*)d_in[3];
    const float* maa_k = (const float*)d_in[4];
    const float* maa_v = (const float*)d_in[5];
    const float* maa_r = (const float*)d_in[6];
    const float* maa_g = (const float*)d_in[7];
    const float* w1    = (const float*)d_in[8];
    const float* w2    = (const float*)d_in[9];
    const float* decay = (const float*)d_in[10];
    const float* dw1   = (const float*)d_in[11];
    const float* dw2   = (const float*)d_in[12];
    const float* faaaa = (const float*)d_in[13];
    const float* Wr    = (const float*)d_in[14];
    const float* Wk    = (const float*)d_in[15];
    const float* Wv    = (const float*)d_in[16];
    const float* Wg    = (const float*)d_in[17];
    const float* Wo    = (const float*)d_in[18];
    const float* gnw   = (const float*)d_in[19];
    const float* gnb   = (const float*)d_in[20];

    float* ws    = (float*)d_ws;
    float* sx    = ws;                    // 65536
    float* xxx   = ws + 65536;            // 65536
    float* T     = ws + 131072;           // 5120
    float* xw    = ws + 136192;           // 65536
    float* T2    = ws + 201728;           // 2048
    float* Wd    = ws + 203776;           // 65536
    float* xbuf  = ws + 269312;           // 4*65536  (xr,xk,xv,xg)
    float* rkvg  = ws + 531456;           // 4*65536  (r,k,v,gpre)
    float* G     = ws + 793600;           // 65536
    float* partA = ws + 859136;           // 16*65536 split-K partials (rkvg)
    float* partY = ws + 1907712;          // 4*65536  split-K partials (y)

    float* y   = (float*)d_out;
    float* nst = y + Bc * Ec;             // new_state follows y

    long n4 = (long)Bc * ROWSc * Ec / 4;  // 25,952,256 float4s
    copy_state<<<2048, 256, 0, stream>>>((const float4*)state, (float4*)nst, n4);
    prep<<<256, 256, 0, stream>>>(x, state, maa_x, sx, xxx, nst);
    mix1<<<Bc, 160, 0, stream>>>(xxx, w1, T);
    mix2<<<dim3(Bc, 8), 256, 0, stream>>>(x, sx, T, w2, maa_w, maa_k, maa_v,
                                          maa_r, maa_g, xw, xbuf);
    decay1<<<Bc, 64, 0, stream>>>(xw, dw1, T2);
    decay2<<<dim3(Bc, 8), 256, 0, stream>>>(T2, dw2, decay, Wd);
    gemm_rkvg_splitk<<<256, 256, 0, stream>>>(xbuf, Wr, Wk, Wv, Wg, partA);
    reduce_k4<<<1024, 256, 0, stream>>>(partA, rkvg);
    att_gn<<<Bc * Hc, 64, 0, stream>>>(rkvg, Wd, faaaa, state, gnw, gnb, nst, G);
    gemm_one_splitk<<<64, 256, 0, stream>>>(G, Wo, partY);
    reduce_k4<<<256, 256, 0, stream>>>(partY, y);
}